// SAGE_47029891891202
// MI455X (gfx1250) — compile-verified
//
#include <hip/hip_runtime.h>

typedef __attribute__((ext_vector_type(2))) float v2f;
typedef __attribute__((ext_vector_type(4))) float v4f;
typedef __attribute__((ext_vector_type(8))) float v8f;

#define F 128   // nfeat == nhid == 128

// ---------------------------------------------------------------- degree
__global__ void sage_deg_kernel(const int* __restrict__ dst, float* __restrict__ deg, int E) {
    int e = blockIdx.x * blockDim.x + threadIdx.x;
    if (e < E) atomicAdd(&deg[dst[e]], 1.0f);
}

__global__ void sage_invdeg_kernel(float* __restrict__ deg, int N) {
    int i = blockIdx.x * blockDim.x + threadIdx.x;
    if (i < N) deg[i] = 1.0f / fmaxf(deg[i], 1.0f);
}

// ---------------------------------------------------------------- 128x128 transpose
// Wt[n][k] = W[k][n]; makes the WMMA B-operand contiguous in K.
__global__ void sage_transpose_kernel(const float* __restrict__ W, float* __restrict__ Wt) {
    int idx = blockIdx.x * blockDim.x + threadIdx.x;   // 16384 threads
    if (idx < F * F) {
        int k = idx >> 7;
        int n = idx & (F - 1);
        Wt[(size_t)n * F + k] = W[idx];
    }
}

// ------------------------------------------------- neighbor scatter-add
// One wave (32 lanes) per edge; each lane moves 4 consecutive floats.
__global__ void sage_scatter_kernel(const float* __restrict__ feat,
                                    const int* __restrict__ src,
                                    const int* __restrict__ dst,
                                    float* __restrict__ neigh, int E) {
    int idx  = blockIdx.x * blockDim.x + threadIdx.x;
    int e    = idx >> 5;
    int lane = idx & 31;
    if (e >= E) return;
    int s = src[e];
    int d = dst[e];
    const float4 v = *(const float4*)(feat + (size_t)s * F + lane * 4);
    float* o = neigh + (size_t)d * F + lane * 4;
    atomicAdd(o + 0, v.x);
    atomicAdd(o + 1, v.y);
    atomicAdd(o + 2, v.z);
    atomicAdd(o + 3, v.w);
}

// ------------------------------------------------- layer 1: fused WMMA GEMM
// h[16 rows x 128 cols] per wave:
//   h = relu( X_tile @ Wself + (Neigh_tile * invdeg) @ Wneigh + b )
// V_WMMA_F32_16X16X4_F32.
// K consumed in a permuted order (legal: A and B use the same permutation):
// chunk g covers K = 8g..8g+7; lane-half 0 holds K=8g..8g+3, half 1 holds
// K=8g+4..8g+7, each as ONE aligned b128 load; (x,y)->step 2g, (z,w)->step 2g+1.
// __launch_bounds__(256, 4): allow up to 256 VGPRs so both staged B arrays
// (64 VGPRs) stay live alongside the 64 accumulator VGPRs -> one load clause
// + one wait + 32 back-to-back WMMAs per chunk.
__global__ __launch_bounds__(256, 4) void sage_layer1_wmma(
        const float* __restrict__ x, const float* __restrict__ neigh,
        const float* __restrict__ invdeg,
        const float* __restrict__ WselfT,   // [n][k] = Wself[k][n]
        const float* __restrict__ WneighT,  // [n][k] = Wneigh[k][n]
        const float* __restrict__ bias,
        float* __restrict__ h, int N, int ntiles) {
    int lane = threadIdx.x & 31;
    int tile = blockIdx.x * (blockDim.x >> 5) + (threadIdx.x >> 5);
    if (tile >= ntiles) return;                 // wave-uniform: EXEC stays all-1s
    int row0 = tile * 16;

    int m   = lane & 15;                        // A row for this lane
    int n   = lane & 15;                        // B/C column within a 16-wide tile
    int kk2 = (lane >> 4) * 4;                  // half-wave K sub-offset (0 or 4)

    int rA = row0 + m;
    if (rA > N - 1) rA = N - 1;                 // clamp (N is a multiple of 16 anyway)
    float inv = invdeg[rA];
    const float* xb  = x       + (size_t)rA * F + kk2;
    const float* nb  = neigh   + (size_t)rA * F + kk2;
    const float* wsb = WselfT  + (size_t)n * F + kk2;   // tile offset t*8192B fits imm24
    const float* wnb = WneighT + (size_t)n * F + kk2;

    v8f acc[8];
#pragma unroll
    for (int t = 0; t < 8; ++t) acc[t] = (v8f){0.f,0.f,0.f,0.f,0.f,0.f,0.f,0.f};

    for (int g = 0; g < F / 8; ++g) {           // 16 chunks of 8 K-values
        int o = g * 8;
        // ---- stage ALL operands first (one load clause, one wait) ----
        v4f aS4 = *(const v4f*)(xb + o);
        v4f aN4 = *(const v4f*)(nb + o);
        v4f bs[8], bn[8];
#pragma unroll
        for (int t = 0; t < 8; ++t) {
            bs[t] = *(const v4f*)(wsb + (size_t)t * 16 * F + o);
            bn[t] = *(const v4f*)(wnb + (size_t)t * 16 * F + o);
        }
        aN4 *= inv;                             // mean aggregation scaling
        v2f aS_lo = (v2f){aS4.x, aS4.y};
        v2f aS_hi = (v2f){aS4.z, aS4.w};
        v2f aN_lo = (v2f){aN4.x, aN4.y};
        v2f aN_hi = (v2f){aN4.z, aN4.w};
        // ---- dense WMMA block: 32 matrix ops, no interleaved loads ----
#pragma unroll
        for (int t = 0; t < 8; ++t) {
            v2f b0 = (v2f){bs[t].x, bs[t].y};
            v2f b1 = (v2f){bn[t].x, bn[t].y};
            acc[t] = __builtin_amdgcn_wmma_f32_16x16x4_f32(
                false, aS_lo, false, b0, (short)0, acc[t], false, false);
            acc[t] = __builtin_amdgcn_wmma_f32_16x16x4_f32(
                false, aN_lo, false, b1, (short)0, acc[t], false, false);
        }
#pragma unroll
        for (int t = 0; t < 8; ++t) {
            v2f b2 = (v2f){bs[t].z, bs[t].w};
            v2f b3 = (v2f){bn[t].z, bn[t].w};
            acc[t] = __builtin_amdgcn_wmma_f32_16x16x4_f32(
                false, aS_hi, false, b2, (short)0, acc[t], false, false);
            acc[t] = __builtin_amdgcn_wmma_f32_16x16x4_f32(
                false, aN_hi, false, b3, (short)0, acc[t], false, false);
        }
    }

    // C/D layout: VGPR j -> row = row0 + (lane>>4)*8 + j, col = t*16 + (lane&15)
    int crow = row0 + (lane >> 4) * 8;
#pragma unroll
    for (int t = 0; t < 8; ++t) {
        float bv = bias[t * 16 + n];
#pragma unroll
        for (int j = 0; j < 8; ++j) {
            int r = crow + j;
            if (r < N) {
                float v = acc[t][j] + bv;
                h[(size_t)r * F + t * 16 + n] = fmaxf(v, 0.0f);
            }
        }
    }
}

// ------------------------------------------------- layer 2: 128->2 + log_softmax
// Output width is 2: WMMA would waste 7/8 of the tile; one wave per node.
__global__ void sage_layer2_kernel(const float* __restrict__ h,
                                   const float* __restrict__ neigh,
                                   const float* __restrict__ invdeg,
                                   const float* __restrict__ Ws2,
                                   const float* __restrict__ Wn2,
                                   const float* __restrict__ b2,
                                   float* __restrict__ out, int N) {
    int idx  = blockIdx.x * blockDim.x + threadIdx.x;
    int node = idx >> 5;
    int lane = idx & 31;
    if (node >= N) return;
    const float* hr = h     + (size_t)node * F;
    const float* nr = neigh + (size_t)node * F;
    float inv = invdeg[node];
    float l0 = 0.f, l1 = 0.f;
#pragma unroll
    for (int f = lane; f < F; f += 32) {
        float hv = hr[f];
        float nv = nr[f] * inv;
        l0 += hv * Ws2[f * 2 + 0] + nv * Wn2[f * 2 + 0];
        l1 += hv * Ws2[f * 2 + 1] + nv * Wn2[f * 2 + 1];
    }
#pragma unroll
    for (int off = 16; off > 0; off >>= 1) {
        l0 += __shfl_down(l0, off, 32);
        l1 += __shfl_down(l1, off, 32);
    }
    if (lane == 0) {
        l0 += b2[0];
        l1 += b2[1];
        float mx  = fmaxf(l0, l1);
        float lse = mx + logf(expf(l0 - mx) + expf(l1 - mx));
        out[(size_t)node * 2 + 0] = l0 - lse;
        out[(size_t)node * 2 + 1] = l1 - lse;
    }
}

extern "C" void kernel_launch(void* const* d_in, const int* in_sizes, int n_in,
                              void* d_out, int out_size, void* d_ws, size_t ws_size,
                              hipStream_t stream) {
    const float* x      = (const float*)d_in[0];
    const int*   src    = (const int*)  d_in[1];
    const int*   dst    = (const int*)  d_in[2];
    const float* Wself1 = (const float*)d_in[3];
    const float* Wneigh1= (const float*)d_in[4];
    const float* b1     = (const float*)d_in[5];
    const float* Ws2    = (const float*)d_in[6];
    const float* Wn2    = (const float*)d_in[7];
    const float* b2     = (const float*)d_in[8];
    float* out = (float*)d_out;

    const int N = in_sizes[0] / F;   // 100000
    const int E = in_sizes[1];       // 1600000

    // workspace layout: [deg | neigh_sum (N*F) | h (N*F) | WselfT | WneighT]
    char* ws = (char*)d_ws;
    size_t degBytes  = ((size_t)N * sizeof(float) + 511) & ~(size_t)511;
    size_t featBytes = (size_t)N * F * sizeof(float);
    size_t wBytes    = (size_t)F * F * sizeof(float);
    float* deg     = (float*)ws;
    float* neigh   = (float*)(ws + degBytes);
    float* h       = (float*)(ws + degBytes + featBytes);
    float* WselfT  = (float*)(ws + degBytes + 2 * featBytes);
    float* WneighT = (float*)(ws + degBytes + 2 * featBytes + wBytes);

    // zero accumulation buffers (graph-capture safe)
    hipMemsetAsync(ws, 0, degBytes + featBytes, stream);

    // transpose layer-1 weights (tiny; amortized over 6250 GEMM waves)
    sage_transpose_kernel<<<(F * F + 255) / 256, 256, 0, stream>>>(Wself1, WselfT);
    sage_transpose_kernel<<<(F * F + 255) / 256, 256, 0, stream>>>(Wneigh1, WneighT);

    unsigned edgeWaveBlocks = (unsigned)(((size_t)E * 32 + 255) / 256);

    sage_deg_kernel<<<(E + 255) / 256, 256, 0, stream>>>(dst, deg, E);
    sage_scatter_kernel<<<edgeWaveBlocks, 256, 0, stream>>>(x, src, dst, neigh, E);
    sage_invdeg_kernel<<<(N + 255) / 256, 256, 0, stream>>>(deg, N);

    int ntiles = (N + 15) / 16;      // 6250
    sage_layer1_wmma<<<(ntiles + 7) / 8, 256, 0, stream>>>(
        x, neigh, deg, WselfT, WneighT, b1, h, N, ntiles);

    hipMemsetAsync(neigh, 0, featBytes, stream);
    sage_scatter_kernel<<<edgeWaveBlocks, 256, 0, stream>>>(h, src, dst, neigh, E);

    unsigned nodeWaveBlocks = (unsigned)(((size_t)N * 32 + 255) / 256);
    sage_layer2_kernel<<<nodeWaveBlocks, 256, 0, stream>>>(
        h, neigh, deg, Ws2, Wn2, b2, out, N);
}